// STDDAGMMLoss_70617852281271
// MI455X (gfx1250) — compile-verified
//
#include <hip/hip_runtime.h>
#include <stdint.h>
#include <math.h>

#define BATCH 262144
#define KMIX  8
#define DFEAT 32

typedef __attribute__((ext_vector_type(2))) float v2f;
typedef __attribute__((ext_vector_type(8))) float v8f;

// ---------------- workspace layout (float offsets) ----------------
#define OFF_GSUM  0        // 8
#define OFF_S1    8        // 256
#define OFF_M2    288      // 8192 (16-float aligned)
#define OFF_SUME  8480
#define OFF_SUMR  8481
#define OFF_SING  8482
#define OFF_CONST 8484     // 8
#define OFF_MU    8496     // 256 (16-float aligned)
#define OFF_LINV  8768     // 8192 (16-float aligned)
#define ZERO_N    8484

// ---------------- CDNA5 helpers ----------------
__device__ __forceinline__ unsigned lds_off32(const void* p) {
  // generic LDS pointer: low 32 bits are the LDS byte offset (ISA aperture rule)
  return (unsigned)(unsigned long long)p;
}
__device__ __forceinline__ void async_b128(unsigned ldsaddr, unsigned voff, const void* base) {
  // GVS mode: SADDR(64) + VADDR(32) -> 16B global->LDS async transfer per lane
  asm volatile("global_load_async_to_lds_b128 %0, %1, %2"
               :: "v"(ldsaddr), "v"(voff), "s"((unsigned long long)base)
               : "memory");
}
__device__ __forceinline__ void wait_async0() {
  asm volatile("s_wait_asynccnt 0" ::: "memory");
}
__device__ __forceinline__ void wait_ds0() {
  asm volatile("s_wait_dscnt 0" ::: "memory");
}

// ---------------- kernel 0: zero accumulators ----------------
__global__ void kzero(float* __restrict__ w) {
  int i = blockIdx.x * blockDim.x + threadIdx.x;
  if (i < ZERO_N) w[i] = 0.f;
}

// ---------------- kernel 1: gsum / S1 / M2 via WMMA f32 16x16x4 ----------------
// wave k (of 8) accumulates M2[k] = sum_b gamma[b,k] * f[b,:] f[b,:]^T
// A tile: A[d, b] = gamma[b,k]*f[b,d] (16x4), B tile: B[b, e] = f[b,e] (4x16)
#define FSTR_A 40   // padded LDS row stride (floats): 160B, 16B aligned, conflict-free
__global__ __launch_bounds__(256) void kstats(const float* __restrict__ gamma,
                                              const float* __restrict__ feat,
                                              float* __restrict__ w) {
  __shared__ __align__(16) float featS[64 * FSTR_A];
  __shared__ __align__(16) float gammaS[64 * 8];
  const int tid  = threadIdx.x;
  const int lane = tid & 31;
  const int k    = tid >> 5;     // wave id == mixture component
  const int h    = lane >> 4;    // lane half
  const int m    = lane & 15;

  v8f c00 = {}, c01 = {}, c10 = {}, c11 = {};
  float s1a0 = 0.f, s1a1 = 0.f, gacc = 0.f;

  const unsigned fbase = lds_off32(featS);
  const unsigned gbase = lds_off32(gammaS);
  const int row_wg = blockIdx.x * 1024;       // 256 WGs * 1024 rows

  for (int ch = 0; ch < 16; ++ch) {           // 16 chunks of 64 rows
    __syncthreads();                           // previous chunk consumed
    const int row0 = row_wg + ch * 64;
    {   // features: 64 rows * 128B = 512 x b128 transfers
      const void* fb = feat + (size_t)row0 * DFEAT;
      int i0 = tid, i1 = tid + 256;
      async_b128(fbase + (unsigned)((i0 >> 3) * (FSTR_A * 4) + (i0 & 7) * 16),
                 (unsigned)(i0 * 16), fb);
      async_b128(fbase + (unsigned)((i1 >> 3) * (FSTR_A * 4) + (i1 & 7) * 16),
                 (unsigned)(i1 * 16), fb);
      if (tid < 128) {  // gamma: 64 rows * 32B = 128 x b128 (wave-uniform branch)
        const void* gb = gamma + (size_t)row0 * KMIX;
        async_b128(gbase + (unsigned)(tid * 16), (unsigned)(tid * 16), gb);
      }
    }
    wait_async0();
    __syncthreads();

    #pragma unroll 4
    for (int s = 0; s < 16; ++s) {            // 4 batch rows / step
      const int r0 = 4 * s + 2 * h;           // K-dim rows for this lane half
      float w0  = gammaS[r0 * 8 + k];
      float w1  = gammaS[(r0 + 1) * 8 + k];
      float f00 = featS[r0 * FSTR_A + m];
      float f01 = featS[(r0 + 1) * FSTR_A + m];
      float f10 = featS[r0 * FSTR_A + 16 + m];
      float f11 = featS[(r0 + 1) * FSTR_A + 16 + m];
      v2f a0 = {w0 * f00, w1 * f01};
      v2f a1 = {w0 * f10, w1 * f11};
      v2f b0 = {f00, f01};
      v2f b1 = {f10, f11};
      c00 = __builtin_amdgcn_wmma_f32_16x16x4_f32(false, a0, false, b0, (short)0, c00, false, false);
      c01 = __builtin_amdgcn_wmma_f32_16x16x4_f32(false, a0, false, b1, (short)0, c01, false, false);
      c10 = __builtin_amdgcn_wmma_f32_16x16x4_f32(false, a1, false, b0, (short)0, c10, false, false);
      c11 = __builtin_amdgcn_wmma_f32_16x16x4_f32(false, a1, false, b1, (short)0, c11, false, false);
      s1a0 += a0.x + a0.y;
      s1a1 += a1.x + a1.y;
      gacc += w0 + w1;
    }
  }

  // flush partials: C/D layout -> lane L, vgpr i holds D[(L>>4)*8+i][L&15]
  float* M2 = w + OFF_M2 + k * 1024;
  #pragma unroll
  for (int i = 0; i < 8; ++i) {
    int d = h * 8 + i;
    atomicAdd(&M2[d * 32 + m],             c00[i]);
    atomicAdd(&M2[d * 32 + 16 + m],        c01[i]);
    atomicAdd(&M2[(16 + d) * 32 + m],      c10[i]);
    atomicAdd(&M2[(16 + d) * 32 + 16 + m], c11[i]);
  }
  atomicAdd(&w[OFF_S1 + k * 32 + m],      s1a0);
  atomicAdd(&w[OFF_S1 + k * 32 + 16 + m], s1a1);
  if (m == 0) atomicAdd(&w[OFF_GSUM + k], gacc);
}

// ---------------- kernel 2: sigma, Cholesky, Linv, logdet, singularity ----------------
__global__ __launch_bounds__(256) void kchol(float* __restrict__ w) {
  __shared__ __align__(16) float S[8][32][33];
  __shared__ float muL[8 * 32];
  __shared__ float red[256];
  const int tid = threadIdx.x;
  const int k   = tid >> 5;   // wave == matrix
  const int i   = tid & 31;   // row

  const float gs    = w[OFF_GSUM + k];
  const float denom = gs + 1e-8f;
  const float s1i   = w[OFF_S1 + k * 32 + i];
  const float mui   = s1i / denom;
  muL[k * 32 + i] = mui;
  w[OFF_MU + k * 32 + i] = mui;
  __syncthreads();

  float diag = 0.f;
  for (int j = 0; j < 32; ++j) {
    float m2  = w[OFF_M2 + k * 1024 + i * 32 + j];
    float muj = muL[k * 32 + j];
    float s1j = w[OFF_S1 + k * 32 + j];
    float sig = (m2 - mui * s1j - s1i * muj + gs * mui * muj) / denom;
    if (i == j) { sig += 1e-6f; diag = sig; }
    S[k][i][j] = sig;
  }
  red[tid] = 1.f / (diag + 1e-8f);
  __syncthreads();
  if (tid == 0) {
    float s = 0.f;
    for (int t = 0; t < 256; ++t) s += red[t];
    w[OFF_SING] = s;
  }

  // in-place Cholesky (lower), thread i owns row i
  for (int j = 0; j < 32; ++j) {
    __syncthreads();
    float ljj = sqrtf(S[k][j][j]);
    float lij = 0.f;
    if (i >= j) { lij = S[k][i][j] / ljj; S[k][i][j] = lij; }
    __syncthreads();
    for (int c = j + 1; c <= i; ++c) S[k][i][c] -= lij * S[k][c][j];
  }
  __syncthreads();

  // logdet + per-cluster constant
  float t = logf(S[k][i][i]);
  for (int msk = 1; msk < 32; msk <<= 1) t += __shfl_xor(t, msk, 32);
  if (i == 0) {
    float logdet = 2.f * t;
    float phi = gs / (float)BATCH;
    w[OFF_CONST + k] = logf(phi + 1e-10f) - 0.5f * (logdet + 32.f * logf(6.28318f));
  }

  // Linv: lane i solves column i by forward substitution; store row-major Linv[r][c]
  {
    float x[32];
    #pragma unroll
    for (int a = 0; a < 32; ++a) x[a] = 0.f;
    for (int r = i; r < 32; ++r) {
      float s = (r == i) ? 1.f : 0.f;
      for (int j2 = i; j2 < r; ++j2) s -= S[k][r][j2] * x[j2];
      x[r] = s / S[k][r][r];
    }
    for (int r = 0; r < 32; ++r)
      w[OFF_LINV + k * 1024 + r * 32 + i] = x[r];
  }
}

// ---------------- kernel 3: energy via WMMA (y = diff * Linv^T) + recon sum ----------------
#define FSTR_C 36   // 144B rows: 16B-aligned, conflict-free for 16-row lane groups
__global__ __launch_bounds__(256) void kenergy(const float* __restrict__ feat,
                                               const float* __restrict__ recon,
                                               float* __restrict__ w) {
  __shared__ __align__(16) float linvS[8 * 32 * FSTR_C];
  __shared__ __align__(16) float muS[256];
  __shared__ __align__(16) float featS[8][16 * FSTR_C];
  const int tid  = threadIdx.x;
  const int lane = tid & 31;
  const int wv   = tid >> 5;
  const int h    = lane >> 4;
  const int m    = lane & 15;

  {   // Linv -> LDS (padded stride), 2048 b128 transfers
    unsigned lb = lds_off32(linvS);
    const void* gb = w + OFF_LINV;
    #pragma unroll
    for (int t = 0; t < 8; ++t) {
      int idx = tid + t * 256;          // 0..2047
      async_b128(lb + (unsigned)((idx >> 3) * (FSTR_C * 4) + (idx & 7) * 16),
                 (unsigned)(idx * 16), gb);
    }
  }
  if (tid < 64) ((float4*)muS)[tid] = ((const float4*)(w + OFF_MU))[tid];

  {   // reconstruction-error partial sum (grid of B/2 threads covers B once, 2 elements each)
    float racc = 0.f;
    int g = blockIdx.x * 256 + tid;
    racc += recon[g] + recon[g + 131072];
    for (int msk = 1; msk < 32; msk <<= 1) racc += __shfl_xor(racc, msk, 32);
    if (lane == 0) atomicAdd(&w[OFF_SUMR], racc);
  }

  float ck[8];
  #pragma unroll
  for (int q = 0; q < 8; ++q) ck[q] = w[OFF_CONST + q];

  wait_async0();
  __syncthreads();

  const int wid = blockIdx.x * 8 + wv;     // 4096 waves, 16384 tiles of 16 rows
  const unsigned fS = lds_off32(&featS[wv][0]);
  float eacc = 0.f;

  for (int t = 0; t < 4; ++t) {
    const int b0 = (wid + t * 4096) * 16;
    wait_ds0();                             // previous tile's LDS reads done
    const void* gb = feat + (size_t)b0 * DFEAT;
    #pragma unroll
    for (int j = 0; j < 4; ++j) {           // 16 rows * 128B = 128 b128, wave-private
      int idx = lane + j * 32;
      async_b128(fS + (unsigned)((idx >> 3) * (FSTR_C * 4) + (idx & 7) * 16),
                 (unsigned)(idx * 16), gb);
    }
    wait_async0();

    float mrun[8], srun[8];
    #pragma unroll
    for (int i2 = 0; i2 < 8; ++i2) { mrun[i2] = -INFINITY; srun[i2] = 0.f; }

    for (int kq = 0; kq < 8; ++kq) {
      v8f c0 = {}, c1 = {};
      #pragma unroll
      for (int s = 0; s < 8; ++s) {         // contraction over d in chunks of 4
        int d0 = 4 * s + 2 * h;
        v2f f   = *(const v2f*)&featS[wv][m * FSTR_C + d0];
        v2f mu2 = *(const v2f*)&muS[kq * 32 + d0];
        v2f a   = f - mu2;                  // A[M=row m][K=d]
        v2f bb0 = *(const v2f*)&linvS[(kq * 32 + m) * FSTR_C + d0];       // B[K=d][N=e]
        v2f bb1 = *(const v2f*)&linvS[(kq * 32 + 16 + m) * FSTR_C + d0];
        c0 = __builtin_amdgcn_wmma_f32_16x16x4_f32(false, a, false, bb0, (short)0, c0, false, false);
        c1 = __builtin_amdgcn_wmma_f32_16x16x4_f32(false, a, false, bb1, (short)0, c1, false, false);
      }
      #pragma unroll
      for (int i2 = 0; i2 < 8; ++i2) {      // mahal = sum_e y^2, reduce over 16 lanes
        float sq = c0[i2] * c0[i2] + c1[i2] * c1[i2];
        sq += __shfl_xor(sq, 1, 32);
        sq += __shfl_xor(sq, 2, 32);
        sq += __shfl_xor(sq, 4, 32);
        sq += __shfl_xor(sq, 8, 32);
        float v  = ck[kq] - 0.5f * sq;      // wlp for row h*8+i2
        float mn = fmaxf(mrun[i2], v);      // branchless online logsumexp
        srun[i2] = srun[i2] * __expf(mrun[i2] - mn) + __expf(v - mn);
        mrun[i2] = mn;
      }
    }
    #pragma unroll
    for (int i2 = 0; i2 < 8; ++i2) {
      float e = -(mrun[i2] + logf(srun[i2] + 1e-10f));
      e = __builtin_isfinite(e) ? e : 0.f;
      eacc += (m == 0) ? e : 0.f;           // lanes 0 & 16 own distinct row halves
    }
  }
  if (m == 0) atomicAdd(&w[OFF_SUME], eacc);
}

// ---------------- kernel 4: assemble scalar ----------------
__global__ void kfinal(const float* __restrict__ w, float* __restrict__ out) {
  float recon_mean  = w[OFF_SUMR] / (float)BATCH;
  float energy_mean = w[OFF_SUME] / (float)BATCH;
  out[0] = recon_mean + 0.1f * energy_mean + 0.005f * w[OFF_SING];
}

// ---------------- host ----------------
extern "C" void kernel_launch(void* const* d_in, const int* in_sizes, int n_in,
                              void* d_out, int out_size, void* d_ws, size_t ws_size,
                              hipStream_t stream) {
  (void)in_sizes; (void)n_in; (void)out_size; (void)ws_size;
  const float* gamma = (const float*)d_in[0];
  const float* feat  = (const float*)d_in[1];
  const float* recon = (const float*)d_in[2];
  float* w   = (float*)d_ws;
  float* out = (float*)d_out;

  kzero  <<<dim3((ZERO_N + 255) / 256), dim3(256), 0, stream>>>(w);
  kstats <<<dim3(256), dim3(256), 0, stream>>>(gamma, feat, w);
  kchol  <<<dim3(1),   dim3(256), 0, stream>>>(w);
  kenergy<<<dim3(512), dim3(256), 0, stream>>>(feat, recon, w);
  kfinal <<<dim3(1),   dim3(1),   0, stream>>>(w, out);
}